// GroupedQueryAttention_80985903333888
// MI455X (gfx1250) — compile-verified
//
#include <hip/hip_runtime.h>
#include <hip/hip_bf16.h>

typedef __attribute__((ext_vector_type(2))) float v2f;
typedef __attribute__((ext_vector_type(8))) float v8f;

#define BLOCK 128

__device__ __forceinline__ v8f wmma4(v2f a, v2f b, v8f c) {
  // D = A(16x4 f32) * B(4x16 f32) + C(16x16 f32)
  return __builtin_amdgcn_wmma_f32_16x16x4_f32(false, a, false, b, (short)0, c,
                                               false, false);
}

// Wq [1024,16,64] -> Wq_g [1024,4,64] (sum over group axis, g inner of h=(kvh,g))
__global__ void reduce_wq_kernel(const float* __restrict__ Wq,
                                 float* __restrict__ Wqg) {
  int idx = blockIdx.x * blockDim.x + threadIdx.x;
  if (idx >= 1024 * 256) return;
  int i = idx >> 8;          // in_dim
  int c = idx & 255;         // kvh*64 + d
  int h = c >> 6, d = c & 63;
  const float* p = Wq + (size_t)i * 1024 + h * 256 + d;
  Wqg[idx] = p[0] + p[64] + p[128] + p[192];
}

// C[M,N] = A[M,K] * B[K,N], fp32, row-major. 64x64 tile per block, 4 waves.
__global__ __launch_bounds__(BLOCK) void gemm_f32_wmma(
    const float* __restrict__ A, const float* __restrict__ B,
    float* __restrict__ C, int M, int N, int K) {
  __shared__ __align__(16) float As[64][18];  // pad: conflict-free A frags
  __shared__ __align__(16) float Bs[16][66];  // pad: conflict-free B frags
  const int tid  = threadIdx.x;
  const int wave = tid >> 5;
  const int lane = tid & 31;
  const int half = lane >> 4;
  const int l16  = lane & 15;
  const int row0 = blockIdx.y * 64;
  const int col0 = blockIdx.x * 64;

  v8f acc[4];
#pragma unroll
  for (int nt = 0; nt < 4; ++nt) acc[nt] = (v8f)(0.0f);

  for (int kk = 0; kk < K; kk += 16) {
    __syncthreads();
    // A tile 64x16
#pragma unroll
    for (int f = tid; f < 256; f += BLOCK) {
      int r = f >> 2, c = (f & 3) << 2;
      const float4 v = *(const float4*)(A + (size_t)(row0 + r) * K + kk + c);
      As[r][c] = v.x; As[r][c + 1] = v.y; As[r][c + 2] = v.z; As[r][c + 3] = v.w;
    }
    // B tile 16x64
#pragma unroll
    for (int f = tid; f < 256; f += BLOCK) {
      int r = f >> 4, c = (f & 15) << 2;
      const float4 v = *(const float4*)(B + (size_t)(kk + r) * N + col0 + c);
      Bs[r][c] = v.x; Bs[r][c + 1] = v.y; Bs[r][c + 2] = v.z; Bs[r][c + 3] = v.w;
    }
    __syncthreads();
#pragma unroll
    for (int k0 = 0; k0 < 16; k0 += 4) {
      // A frag: lane holds M=l16 (+wave*16), K = k0 + 2*half + {0,1}
      v2f a = *(const v2f*)&As[wave * 16 + l16][k0 + 2 * half];
#pragma unroll
      for (int nt = 0; nt < 4; ++nt) {
        v2f b;
        b.x = Bs[k0 + 2 * half + 0][nt * 16 + l16];
        b.y = Bs[k0 + 2 * half + 1][nt * 16 + l16];
        acc[nt] = wmma4(a, b, acc[nt]);
      }
    }
  }
  // C/D layout: VGPR r -> row r + 8*half, col = l16
#pragma unroll
  for (int nt = 0; nt < 4; ++nt)
#pragma unroll
    for (int r = 0; r < 8; ++r)
      C[(size_t)(row0 + wave * 16 + r + 8 * half) * N + col0 + nt * 16 + l16] =
          acc[nt][r];
}

// Flash attention, causal, per (b, kv-head). Qg/Kp/Vp/Op are [B*S, 256],
// head h occupies columns [h*64, h*64+64). Block = 64 query rows, 4 waves.
__global__ __launch_bounds__(BLOCK) void fa_f32_wmma(
    const float* __restrict__ Qg, const float* __restrict__ Kp,
    const float* __restrict__ Vp, float* __restrict__ Op, int S) {
  __shared__ __align__(16) float Ks[16][66];
  __shared__ __align__(16) float Vs[16][66];
  __shared__ __align__(16) float Ps[4][16][18];
  const int tid  = threadIdx.x;
  const int wave = tid >> 5;
  const int lane = tid & 31;
  const int half = lane >> 4;
  const int l16  = lane & 15;
  const int q0   = blockIdx.x * 64;
  const int b    = blockIdx.y >> 2;
  const int h    = blockIdx.y & 3;

  // Q fragments in registers: qf[i] covers d = i*4 + 2*half + {0,1}
  v2f qf[16];
  {
    const int qrow = q0 + wave * 16 + l16;
    const float* qptr =
        Qg + (size_t)(b * S + qrow) * 256 + h * 64 + 2 * half;
#pragma unroll
    for (int i = 0; i < 16; ++i) qf[i] = *(const v2f*)(qptr + i * 4);
  }

  v8f oacc[4];
#pragma unroll
  for (int nt = 0; nt < 4; ++nt) oacc[nt] = (v8f)(0.0f);
  float m[8], l[8];
#pragma unroll
  for (int r = 0; r < 8; ++r) { m[r] = -1e30f; l[r] = 0.0f; }

  const int ntiles = (q0 >> 4) + 4;  // keys up to q0+63 (causal)
  for (int j = 0; j < ntiles; ++j) {
    const int a0 = j << 4;
    __syncthreads();
#pragma unroll
    for (int f = tid; f < 256; f += BLOCK) {
      int r = f >> 4, c = (f & 15) << 2;
      const size_t g = (size_t)(b * S + a0 + r) * 256 + h * 64 + c;
      const float4 kv = *(const float4*)(Kp + g);
      Ks[r][c] = kv.x; Ks[r][c + 1] = kv.y; Ks[r][c + 2] = kv.z; Ks[r][c + 3] = kv.w;
      const float4 vv = *(const float4*)(Vp + g);
      Vs[r][c] = vv.x; Vs[r][c + 1] = vv.y; Vs[r][c + 2] = vv.z; Vs[r][c + 3] = vv.w;
    }
    __syncthreads();

    // scores S = Q * K^T : B[d][a] = K[a][d] -> Ks[l16][d0+2*half+{0,1}]
    v8f s = (v8f)(0.0f);
#pragma unroll
    for (int i = 0; i < 16; ++i) {
      v2f bk;
      bk.x = Ks[l16][i * 4 + 2 * half + 0];
      bk.y = Ks[l16][i * 4 + 2 * half + 1];
      s = wmma4(qf[i], bk, s);
    }

    // online softmax: element r is row (q0 + wave*16 + r + 8*half), col a0+l16
    const int acol = a0 + l16;
    float pr[8];
#pragma unroll
    for (int r = 0; r < 8; ++r) {
      const int qrow = q0 + wave * 16 + r + 8 * half;
      float x = s[r] * 0.125f;                 // 1/sqrt(64)
      if (acol > qrow) x = -1e30f;             // causal mask
      float mx = x;
#pragma unroll
      for (int off = 1; off < 16; off <<= 1)
        mx = fmaxf(mx, __shfl_xor(mx, off, 16));
      const float mnew  = fmaxf(m[r], mx);
      const float p     = __expf(x - mnew);
      float sum = p;
#pragma unroll
      for (int off = 1; off < 16; off <<= 1) sum += __shfl_xor(sum, off, 16);
      const float scale = __expf(m[r] - mnew);
      l[r] = l[r] * scale + sum;
      m[r] = mnew;
#pragma unroll
      for (int nt = 0; nt < 4; ++nt) oacc[nt][r] *= scale;
      pr[r] = p;
    }

    // P tile -> LDS (row-major [q][a]) so we can re-fragment as WMMA A
#pragma unroll
    for (int r = 0; r < 8; ++r) Ps[wave][r + 8 * half][l16] = pr[r];
    __syncthreads();

    // O += P * V
#pragma unroll
    for (int k0i = 0; k0i < 4; ++k0i) {
      const int k0 = k0i * 4;
      v2f a = *(const v2f*)&Ps[wave][l16][k0 + 2 * half];
#pragma unroll
      for (int nt = 0; nt < 4; ++nt) {
        v2f bv;
        bv.x = Vs[k0 + 2 * half + 0][nt * 16 + l16];
        bv.y = Vs[k0 + 2 * half + 1][nt * 16 + l16];
        oacc[nt] = wmma4(a, bv, oacc[nt]);
      }
    }
  }

  // normalize and store
#pragma unroll
  for (int r = 0; r < 8; ++r) {
    const float inv = 1.0f / l[r];
    const int row = b * S + q0 + wave * 16 + r + 8 * half;
#pragma unroll
    for (int nt = 0; nt < 4; ++nt)
      Op[(size_t)row * 256 + h * 64 + nt * 16 + l16] = oacc[nt][r] * inv;
  }
}

extern "C" void kernel_launch(void* const* d_in, const int* in_sizes, int n_in,
                              void* d_out, int out_size, void* d_ws,
                              size_t ws_size, hipStream_t stream) {
  const float* q_in  = (const float*)d_in[0];
  const float* kv_in = (const float*)d_in[1];
  // d_in[2] = causal mask (tril) — applied analytically in fa_f32_wmma
  const float* Wq = (const float*)d_in[3];
  const float* Wk = (const float*)d_in[4];  // [1024,4,64] == [1024,256]
  const float* Wv = (const float*)d_in[5];
  const float* Wo = (const float*)d_in[6];  // [256,1024]
  float* out = (float*)d_out;

  const int B = 2, S = 2048, IN = 1024;
  const int M = B * S;  // 4096

  float* ws  = (float*)d_ws;
  float* wqg = ws;                    // 1024*256
  float* qg  = wqg + 1024 * 256;      // 4096*256
  float* kp  = qg + M * 256;          // 4096*256
  float* vp  = kp + M * 256;          // 4096*256
  float* op  = vp + M * 256;          // 4096*256  (total ~17.8 MB)

  reduce_wq_kernel<<<(1024 * 256) / 256, 256, 0, stream>>>(Wq, wqg);

  dim3 blk(BLOCK);
  gemm_f32_wmma<<<dim3(256 / 64, M / 64), blk, 0, stream>>>(q_in, wqg, qg, M, 256, IN);
  gemm_f32_wmma<<<dim3(256 / 64, M / 64), blk, 0, stream>>>(kv_in, Wk, kp, M, 256, IN);
  gemm_f32_wmma<<<dim3(256 / 64, M / 64), blk, 0, stream>>>(kv_in, Wv, vp, M, 256, IN);

  fa_f32_wmma<<<dim3(S / 64, B * 4), blk, 0, stream>>>(qg, kp, vp, op, S);

  gemm_f32_wmma<<<dim3(IN / 64, M / 64), blk, 0, stream>>>(op, Wo, out, M, IN, 256);
}